// AdvancedGATModel_22290880266886
// MI455X (gfx1250) — compile-verified
//
#include <hip/hip_runtime.h>
#include <math.h>

// ---------------------------------------------------------------------------
// Problem constants (from reference)
// ---------------------------------------------------------------------------
#define NN 50000      // nodes
#define NE 800000     // edges
#define NG 512        // graphs
#define DD 256        // hidden dim
#define EDF 4         // edge feature dim
#define GDF 13        // global feature dim
#define K0PAD 16      // layer-0 K (14) zero-padded to 16

#define GEMM_WAVES 5            // 3125 M-tiles = 625 blocks * 5 waves (exact)
#define GEMM_MBLK 625
#define GEMM_THREADS (GEMM_WAVES * 32)

typedef float v2f __attribute__((ext_vector_type(2)));
typedef float v8f __attribute__((ext_vector_type(8)));

// ordered-uint encoding for float atomicMax (handles negatives)
static __device__ __forceinline__ unsigned f2ord(float f) {
  unsigned u = __float_as_uint(f);
  return (u & 0x80000000u) ? ~u : (u | 0x80000000u);
}
static __device__ __forceinline__ float ord2f(unsigned u) {
  unsigned b = (u & 0x80000000u) ? (u & 0x7fffffffu) : ~u;
  return __uint_as_float(b);
}
#define ORD_NEG_INF 0x007FFFFFu   // f2ord(-inf)

static __device__ __forceinline__ float sigf(float x) { return 1.0f / (1.0f + expf(-x)); }

// ---------------------------------------------------------------------------
// Fill / pad kernels
// ---------------------------------------------------------------------------
__global__ void fill_f32(float* p, float v, long long n) {
  long long i = (long long)blockIdx.x * blockDim.x + threadIdx.x;
  if (i < n) p[i] = v;
}
__global__ void fill_u32(unsigned* p, unsigned v, long long n) {
  long long i = (long long)blockIdx.x * blockDim.x + threadIdx.x;
  if (i < n) p[i] = v;
}
// x [NN,14] -> xpad [NN,16] (zero-padded K) so the GEMM needs no K guards
__global__ void pad_x(const float* __restrict__ x, float* __restrict__ xp) {
  long long gid = (long long)blockIdx.x * blockDim.x + threadIdx.x;
  if (gid >= (long long)NN * K0PAD) return;
  int c = (int)(gid & (K0PAD - 1));
  long long n = gid >> 4;
  xp[gid] = (c < 14) ? x[n * 14 + c] : 0.0f;
}

// ---------------------------------------------------------------------------
// WMMA f32 GEMM: C[M,Nc] = A[M,K] @ B[K,Nc], row-major.
// Preconditions: M = 50000 (3125 16-row tiles), K % 4 == 0, K <= 256, Nc = 256.
// Block = 5 waves; block stages the K x 64 B-panel in LDS (as row-pair float2s
// so the inner loop reads ds_load_b64), then each wave computes a 16x64 C tile
// with 4 accumulators via V_WMMA_F32_16X16X4_F32.
//   A 16x4 layout : lanes 0-15 M=lane; VGPR0/1 = K{0,1} (sel=0) / K{2,3} (sel=1)
//   B 4x16 layout : lane holds N=lane&15; rows K = 2*sel, 2*sel+1
//   C 16x16 layout: VGPR i -> M = i + 8*sel, N = lane&15
// ---------------------------------------------------------------------------
__global__ __launch_bounds__(GEMM_THREADS)
void wmma_gemm_f32(const float* __restrict__ A, const float* __restrict__ B,
                   float* __restrict__ C, int K, int Nc) {
  // Bs[kp][c] = float2{ B[2*kp][col0+c], B[2*kp+1][col0+c] }, kp < K/2, c < 64
  __shared__ float Bs[(DD / 2) * 64 * 2];          // 64 KB
  const int tid  = threadIdx.x;
  const int wid  = tid >> 5;
  const int lane = tid & 31;
  const int mblk = blockIdx.x % GEMM_MBLK;
  const int tn   = blockIdx.x / GEMM_MBLK;
  const int col0 = tn << 6;

  // cooperative LDS fill of the B panel (coalesced along columns)
  for (int idx = tid; idx < K * 64; idx += GEMM_THREADS) {
    int k = idx >> 6;
    int c = idx & 63;
    Bs[((k >> 1) * 64 + c) * 2 + (k & 1)] = B[(long long)k * Nc + col0 + c];
  }
  __syncthreads();

  const int tm   = mblk * GEMM_WAVES + wid;
  const int row0 = tm << 4;
  const int mi   = lane & 15;
  const int sel  = lane >> 4;                      // 0 or 1
  const float* Arow = A + (long long)(row0 + mi) * K;
  const v2f* Bsv = (const v2f*)Bs;

  v8f acc0 = {}, acc1 = {}, acc2 = {}, acc3 = {};
#pragma unroll 4
  for (int k0 = 0; k0 < K; k0 += 4) {
    v2f av = *(const v2f*)(Arow + k0 + 2 * sel);   // global_load_b64
    const v2f* bp = Bsv + ((k0 >> 1) + sel) * 64 + mi;
    v2f bv0 = bp[0];                               // ds_load_b64 x4
    v2f bv1 = bp[16];
    v2f bv2 = bp[32];
    v2f bv3 = bp[48];
    acc0 = __builtin_amdgcn_wmma_f32_16x16x4_f32(false, av, false, bv0, (short)0, acc0, false, false);
    acc1 = __builtin_amdgcn_wmma_f32_16x16x4_f32(false, av, false, bv1, (short)0, acc1, false, false);
    acc2 = __builtin_amdgcn_wmma_f32_16x16x4_f32(false, av, false, bv2, (short)0, acc2, false, false);
    acc3 = __builtin_amdgcn_wmma_f32_16x16x4_f32(false, av, false, bv3, (short)0, acc3, false, false);
  }
  float* Cr = C + (long long)(row0 + 8 * sel) * Nc + col0 + mi;
#pragma unroll
  for (int i = 0; i < 8; ++i) {
    float* cp = Cr + (long long)i * Nc;
    cp[0]  = acc0[i];
    cp[16] = acc1[i];
    cp[32] = acc2[i];
    cp[48] = acc3[i];
  }
}

// ---------------------------------------------------------------------------
// GAT helper kernels
// ---------------------------------------------------------------------------

// deg[dst]+=1; loop_attr[dst]+=edge_attr (self-loop attr = mean of incoming)
__global__ void edge_deg(const int* __restrict__ dst, const float* __restrict__ ea,
                         float* deg, float* loopA) {
  int e = blockIdx.x * blockDim.x + threadIdx.x;
  if (e >= NE) return;
  int d = dst[e];
  atomicAdd(&deg[d], 1.0f);
#pragma unroll
  for (int j = 0; j < EDF; ++j) atomicAdd(&loopA[(long long)d * EDF + j], ea[(long long)e * EDF + j]);
}
__global__ void loop_div(float* loopA, const float* __restrict__ deg) {
  long long gid = (long long)blockIdx.x * blockDim.x + threadIdx.x;
  if (gid >= (long long)NN * EDF) return;
  float dg = deg[gid / EDF];
  loopA[gid] /= (dg > 1.0f ? dg : 1.0f);
}

// v_e[j*H+h] = sum_c We[j, h*C+c] * aedge[h*C+c]   (collapses edge projection)
__global__ void compute_ve(const float* __restrict__ We, const float* __restrict__ aedge,
                           float* ve, int H) {
  int gid = blockIdx.x * blockDim.x + threadIdx.x;
  if (gid >= EDF * H) return;
  int hh = gid % H, j = gid / H;
  int C = DD / H;
  float s = 0.0f;
  for (int c = 0; c < C; ++c) s += We[(long long)j * DD + hh * C + c] * aedge[hh * C + c];
  ve[j * H + hh] = s;
}

// per-node attention coefficients a_src, a_dst
__global__ void node_attn(const float* __restrict__ hlin, const float* __restrict__ asrc,
                          const float* __restrict__ adst, float* a_s, float* a_d, int H) {
  long long gid = (long long)blockIdx.x * blockDim.x + threadIdx.x;
  if (gid >= (long long)NN * H) return;
  int hh = (int)(gid % H);
  long long n = gid / H;
  int C = DD / H;
  const float* hp = hlin + n * DD + hh * C;
  float ss = 0.0f, sd = 0.0f;
  for (int c = 0; c < C; ++c) { ss += hp[c] * asrc[hh * C + c]; sd += hp[c] * adst[hh * C + c]; }
  a_s[gid] = ss;
  a_d[gid] = sd;
}

// alpha = leaky_relu(a_src[s]+a_dst[d]+ea.v_e, 0.2); segment max via ordered-uint
__global__ void edge_alpha(const int* __restrict__ src, const int* __restrict__ dst,
                           const float* __restrict__ ea, const float* __restrict__ loopA,
                           const float* __restrict__ a_s, const float* __restrict__ a_d,
                           const float* __restrict__ ve, float* alpha, unsigned* segmax, int H) {
  long long gid = (long long)blockIdx.x * blockDim.x + threadIdx.x;
  long long total = (long long)(NE + NN) * H;
  if (gid >= total) return;
  int hh = (int)(gid % H);
  long long i = gid / H;
  int s, d;
  const float* eap;
  if (i < NE) { s = src[i]; d = dst[i]; eap = ea + i * EDF; }
  else        { s = d = (int)(i - NE);  eap = loopA + (long long)s * EDF; }
  float al = a_s[(long long)s * H + hh] + a_d[(long long)d * H + hh];
#pragma unroll
  for (int j = 0; j < EDF; ++j) al += eap[j] * ve[j * H + hh];
  al = al > 0.0f ? al : 0.2f * al;
  alpha[gid] = al;
  atomicMax(&segmax[(long long)d * H + hh], f2ord(al));
}

// alpha <- exp(alpha - max); segment sum
__global__ void edge_exp(const int* __restrict__ dst, const unsigned* __restrict__ segmax,
                         float* alpha, float* segsum, int H) {
  long long gid = (long long)blockIdx.x * blockDim.x + threadIdx.x;
  long long total = (long long)(NE + NN) * H;
  if (gid >= total) return;
  int hh = (int)(gid % H);
  long long i = gid / H;
  int d = (i < NE) ? dst[i] : (int)(i - NE);
  float m = ord2f(segmax[(long long)d * H + hh]);
  float ex = expf(alpha[gid] - m);
  alpha[gid] = ex;
  atomicAdd(&segsum[(long long)d * H + hh], ex);
}

// out[dst, :] += w * hlin[src, :]  (thread = edge x 4-channel chunk)
__global__ void edge_scatter(const int* __restrict__ src, const int* __restrict__ dst,
                             const float* __restrict__ alpha, const float* __restrict__ segsum,
                             const float* __restrict__ hlin, float* outb, int H) {
  long long gid = (long long)blockIdx.x * blockDim.x + threadIdx.x;
  long long total = (long long)(NE + NN) * (DD / 4);
  if (gid >= total) return;
  long long i = gid >> 6;                 // DD/4 == 64 chunks per edge
  int chunk = (int)(gid & 63);
  int s, d;
  if (i < NE) { s = src[i]; d = dst[i]; } else { s = d = (int)(i - NE); }
  int C = DD / H;
  int ch0 = chunk * 4;
  int hh = ch0 / C;                       // C >= 32, so one head per chunk
  float w = alpha[i * H + hh] / (segsum[(long long)d * H + hh] + 1e-16f);
  const float4 hv = *(const float4*)(hlin + (long long)s * DD + ch0);
  float* op = outb + (long long)d * DD + ch0;
  atomicAdd(&op[0], w * hv.x);
  atomicAdd(&op[1], w * hv.y);
  atomicAdd(&op[2], w * hv.z);
  atomicAdd(&op[3], w * hv.w);
}

// h_next = [h_in +] elu(out + b)
__global__ void gat_finish(const float* __restrict__ outb, const float* __restrict__ bias,
                           const float* __restrict__ h_in, float* h_out, int residual) {
  long long gid = (long long)blockIdx.x * blockDim.x + threadIdx.x;
  if (gid >= (long long)NN * DD) return;
  int ch = (int)(gid % DD);
  float v = outb[gid] + bias[ch];
  v = v > 0.0f ? v : (expf(v) - 1.0f);
  h_out[gid] = residual ? (h_in[gid] + v) : v;
}

// ---------------------------------------------------------------------------
// Set2Set kernels (G=512, d=256)
// ---------------------------------------------------------------------------
__global__ void s2s_gates(const float* __restrict__ q_star, const float* __restrict__ hs,
                          const float* __restrict__ Wih, const float* __restrict__ Whh,
                          const float* __restrict__ bih, const float* __restrict__ bhh,
                          float* gates) {
  long long gid = (long long)blockIdx.x * blockDim.x + threadIdx.x;
  if (gid >= (long long)NG * 4 * DD) return;
  int j = (int)(gid % (4 * DD));
  int g = (int)(gid / (4 * DD));
  float s = bih[j] + bhh[j];
  const float* qp = q_star + (long long)g * (2 * DD);
  const float* wi = Wih + (long long)j * (2 * DD);
  for (int k = 0; k < 2 * DD; ++k) s += qp[k] * wi[k];
  const float* hp = hs + (long long)g * DD;
  const float* wh = Whh + (long long)j * DD;
  for (int k = 0; k < DD; ++k) s += hp[k] * wh[k];
  gates[gid] = s;
}
__global__ void s2s_lstm(const float* __restrict__ gates, float* hs, float* cs) {
  long long gid = (long long)blockIdx.x * blockDim.x + threadIdx.x;
  if (gid >= (long long)NG * DD) return;
  int k = (int)(gid % DD);
  int g = (int)(gid / DD);
  const float* gp = gates + (long long)g * (4 * DD);
  float i_ = sigf(gp[k]);
  float f_ = sigf(gp[DD + k]);
  float g_ = tanhf(gp[2 * DD + k]);
  float o_ = sigf(gp[3 * DD + k]);
  float c = f_ * cs[gid] + i_ * g_;
  cs[gid] = c;
  hs[gid] = o_ * tanhf(c);
}
__global__ void s2s_e(const float* __restrict__ xn, const float* __restrict__ hs,
                      const int* __restrict__ batch, float* e_n, unsigned* m_g) {
  int n = blockIdx.x * blockDim.x + threadIdx.x;
  if (n >= NN) return;
  int b = batch[n];
  const float* xp = xn + (long long)n * DD;
  const float* hp = hs + (long long)b * DD;
  float s = 0.0f;
  for (int c = 0; c < DD; ++c) s += xp[c] * hp[c];
  e_n[n] = s;
  atomicMax(&m_g[b], f2ord(s));
}
__global__ void s2s_exp(const int* __restrict__ batch, const unsigned* __restrict__ m_g,
                        float* e_n, float* s_g) {
  int n = blockIdx.x * blockDim.x + threadIdx.x;
  if (n >= NN) return;
  int b = batch[n];
  float m = ord2f(m_g[b]);
  if (!isfinite(m)) m = 0.0f;            // jnp.where(isfinite(m), m, 0)
  float ex = expf(e_n[n] - m);
  e_n[n] = ex;
  atomicAdd(&s_g[b], ex);
}
__global__ void s2s_r(const float* __restrict__ xn, const int* __restrict__ batch,
                      const float* __restrict__ e_n, const float* __restrict__ s_g, float* r) {
  long long gid = (long long)blockIdx.x * blockDim.x + threadIdx.x;
  if (gid >= (long long)NN * (DD / 4)) return;
  long long n = gid >> 6;
  int ch0 = (int)(gid & 63) * 4;
  int b = batch[n];
  float a = e_n[n] / (s_g[b] + 1e-16f);
  const float4 xv = *(const float4*)(xn + n * DD + ch0);
  float* rp = r + (long long)b * DD + ch0;
  atomicAdd(&rp[0], a * xv.x);
  atomicAdd(&rp[1], a * xv.y);
  atomicAdd(&rp[2], a * xv.z);
  atomicAdd(&rp[3], a * xv.w);
}
__global__ void s2s_qstar(const float* __restrict__ hs, const float* __restrict__ r, float* q_star) {
  long long gid = (long long)blockIdx.x * blockDim.x + threadIdx.x;
  if (gid >= (long long)NG * 2 * DD) return;
  int k = (int)(gid % (2 * DD));
  int g = (int)(gid / (2 * DD));
  q_star[gid] = (k < DD) ? hs[(long long)g * DD + k] : r[(long long)g * DD + (k - DD)];
}

// ---------------------------------------------------------------------------
// MLP head
// ---------------------------------------------------------------------------
__global__ void concat_z(const float* __restrict__ q_star, const float* __restrict__ gf, float* z) {
  long long gid = (long long)blockIdx.x * blockDim.x + threadIdx.x;
  const int ZK = 2 * DD + GDF;
  if (gid >= (long long)NG * ZK) return;
  int k = (int)(gid % ZK);
  int g = (int)(gid / ZK);
  z[gid] = (k < 2 * DD) ? q_star[(long long)g * (2 * DD) + k] : gf[(long long)g * GDF + (k - 2 * DD)];
}
__global__ void dense(const float* __restrict__ X, const float* __restrict__ W,
                      const float* __restrict__ b, float* Y, int M, int K, int Nc, int relu) {
  long long gid = (long long)blockIdx.x * blockDim.x + threadIdx.x;
  if (gid >= (long long)M * Nc) return;
  int j = (int)(gid % Nc);
  long long m = gid / Nc;
  const float* xp = X + m * K;
  float s = b[j];
  for (int k = 0; k < K; ++k) s += xp[k] * W[(long long)k * Nc + j];
  Y[gid] = relu ? fmaxf(s, 0.0f) : s;
}

// ---------------------------------------------------------------------------
// Host orchestration
// ---------------------------------------------------------------------------
static inline int cdiv(long long a, long long b) { return (int)((a + b - 1) / b); }

extern "C" void kernel_launch(void* const* d_in, const int* in_sizes, int n_in,
                              void* d_out, int out_size, void* d_ws, size_t ws_size,
                              hipStream_t stream) {
  (void)in_sizes; (void)n_in; (void)out_size; (void)ws_size;
  const float* x     = (const float*)d_in[0];
  const int*   eidx  = (const int*)d_in[1];
  const int*   srcI  = eidx;
  const int*   dstI  = eidx + NE;
  const float* ea    = (const float*)d_in[2];
  const int*   batch = (const int*)d_in[3];
  const float* gf    = (const float*)d_in[4];

  struct Layer { const float *W, *We, *as, *ad, *ae, *b; };
  Layer L[4];
  for (int i = 0; i < 4; ++i) {
    int base = 5 + 6 * i;
    L[i].W  = (const float*)d_in[base + 0];
    L[i].We = (const float*)d_in[base + 1];
    L[i].as = (const float*)d_in[base + 2];
    L[i].ad = (const float*)d_in[base + 3];
    L[i].ae = (const float*)d_in[base + 4];
    L[i].b  = (const float*)d_in[base + 5];
  }
  const float* Wih = (const float*)d_in[29];
  const float* Whh = (const float*)d_in[30];
  const float* bih = (const float*)d_in[31];
  const float* bhh = (const float*)d_in[32];
  const float* p1W = (const float*)d_in[33]; const float* p1b = (const float*)d_in[34];
  const float* p2W = (const float*)d_in[35]; const float* p2b = (const float*)d_in[36];
  const float* p3W = (const float*)d_in[37]; const float* p3b = (const float*)d_in[38];

  // ---- workspace carve-up (floats) ----
  float* ws = (float*)d_ws;
  long long off = 0;
  float* bufA = ws + off; off += (long long)NN * DD;
  float* bufB = ws + off; off += (long long)NN * DD;
  float* bufC = ws + off; off += (long long)NN * DD;
  float* xpad  = ws + off; off += (long long)NN * K0PAD;
  float* deg   = ws + off; off += NN;
  float* loopA = ws + off; off += (long long)NN * EDF;
  float* a_s   = ws + off; off += (long long)NN * 8;
  float* a_d   = ws + off; off += (long long)NN * 8;
  float* ve    = ws + off; off += EDF * 8;
  float* alpha = ws + off; off += (long long)(NE + NN) * 8;
  unsigned* segmax = (unsigned*)(ws + off); off += (long long)NN * 8;
  float* segsum = ws + off; off += (long long)NN * 8;
  float* q_star = ws + off; off += (long long)NG * 2 * DD;
  float* hs     = ws + off; off += (long long)NG * DD;
  float* cs     = ws + off; off += (long long)NG * DD;
  float* gates  = ws + off; off += (long long)NG * 4 * DD;
  float* e_n    = ws + off; off += NN;
  unsigned* m_g = (unsigned*)(ws + off); off += NG;
  float* s_g    = ws + off; off += NG;
  float* rbuf   = ws + off; off += (long long)NG * DD;
  float* zbuf   = ws + off; off += (long long)NG * (2 * DD + GDF);
  float* z1     = ws + off; off += (long long)NG * DD;
  float* z2     = ws + off; off += (long long)NG * (DD / 2);

  const int T = 256;

  // Zero-pad x once: [NN,14] -> [NN,16]
  pad_x<<<cdiv((long long)NN * K0PAD, T), T, 0, stream>>>(x, xpad);

  // ---- 4 GAT layers with triple-buffer rotation ----
  const float* lin_in[4] = { xpad, bufC, bufA, bufB };
  float*       lin_ot[4] = { bufA, bufA, bufB, bufA };
  float*       out_b [4] = { bufB, bufB, bufC, bufC };
  float*       h_nxt [4] = { bufC, bufA, bufB, bufA };
  const int    resid [4] = { 0, 1, 1, 1 };
  const int    Ks    [4] = { K0PAD, DD, DD, DD };
  const int    Hs    [4] = { 8, 8, 8, 1 };

  for (int l = 0; l < 4; ++l) {
    const int H = Hs[l];
    // init scratch
    fill_f32<<<cdiv(NN, T), T, 0, stream>>>(deg, 0.0f, NN);
    fill_f32<<<cdiv((long long)NN * EDF, T), T, 0, stream>>>(loopA, 0.0f, (long long)NN * EDF);
    fill_u32<<<cdiv((long long)NN * 8, T), T, 0, stream>>>(segmax, ORD_NEG_INF, (long long)NN * 8);
    fill_f32<<<cdiv((long long)NN * 8, T), T, 0, stream>>>(segsum, 0.0f, (long long)NN * 8);
    fill_f32<<<cdiv((long long)NN * DD, T), T, 0, stream>>>(out_b[l], 0.0f, (long long)NN * DD);

    // self-loop attr = mean incoming edge attr
    edge_deg<<<cdiv(NE, T), T, 0, stream>>>(dstI, ea, deg, loopA);
    loop_div<<<cdiv((long long)NN * EDF, T), T, 0, stream>>>(loopA, deg);

    // h_lin = h @ W   (WMMA GEMM: 5 waves/block, LDS-staged B panel)
    {
      int blocks = GEMM_MBLK * (DD / 64);   // 625 * 4
      wmma_gemm_f32<<<blocks, GEMM_THREADS, 0, stream>>>(lin_in[l], L[l].W, lin_ot[l], Ks[l], DD);
    }

    // attention coefficients
    compute_ve<<<1, EDF * 8, 0, stream>>>(L[l].We, L[l].ae, ve, H);
    node_attn<<<cdiv((long long)NN * H, T), T, 0, stream>>>(lin_ot[l], L[l].as, L[l].ad, a_s, a_d, H);

    long long eh = (long long)(NE + NN) * H;
    edge_alpha<<<cdiv(eh, T), T, 0, stream>>>(srcI, dstI, ea, loopA, a_s, a_d, ve, alpha, segmax, H);
    edge_exp<<<cdiv(eh, T), T, 0, stream>>>(dstI, segmax, alpha, segsum, H);

    long long sc = (long long)(NE + NN) * (DD / 4);
    edge_scatter<<<cdiv(sc, T), T, 0, stream>>>(srcI, dstI, alpha, segsum, lin_ot[l], out_b[l], H);

    gat_finish<<<cdiv((long long)NN * DD, T), T, 0, stream>>>(out_b[l], L[l].b, lin_in[l], h_nxt[l], resid[l]);
  }
  float* xn = h_nxt[3];   // final node features [NN, DD] (== bufA)

  // ---- Set2Set (3 steps) ----
  fill_f32<<<cdiv((long long)NG * 2 * DD, T), T, 0, stream>>>(q_star, 0.0f, (long long)NG * 2 * DD);
  fill_f32<<<cdiv((long long)NG * DD, T), T, 0, stream>>>(hs, 0.0f, (long long)NG * DD);
  fill_f32<<<cdiv((long long)NG * DD, T), T, 0, stream>>>(cs, 0.0f, (long long)NG * DD);
  for (int step = 0; step < 3; ++step) {
    s2s_gates<<<cdiv((long long)NG * 4 * DD, T), T, 0, stream>>>(q_star, hs, Wih, Whh, bih, bhh, gates);
    s2s_lstm<<<cdiv((long long)NG * DD, T), T, 0, stream>>>(gates, hs, cs);
    fill_u32<<<cdiv(NG, T), T, 0, stream>>>(m_g, ORD_NEG_INF, NG);
    fill_f32<<<cdiv(NG, T), T, 0, stream>>>(s_g, 0.0f, NG);
    fill_f32<<<cdiv((long long)NG * DD, T), T, 0, stream>>>(rbuf, 0.0f, (long long)NG * DD);
    s2s_e<<<cdiv(NN, T), T, 0, stream>>>(xn, hs, batch, e_n, m_g);
    s2s_exp<<<cdiv(NN, T), T, 0, stream>>>(batch, m_g, e_n, s_g);
    s2s_r<<<cdiv((long long)NN * (DD / 4), T), T, 0, stream>>>(xn, batch, e_n, s_g, rbuf);
    s2s_qstar<<<cdiv((long long)NG * 2 * DD, T), T, 0, stream>>>(hs, rbuf, q_star);
  }

  // ---- MLP head ----
  const int ZK = 2 * DD + GDF;  // 525
  concat_z<<<cdiv((long long)NG * ZK, T), T, 0, stream>>>(q_star, gf, zbuf);
  dense<<<cdiv((long long)NG * DD, T), T, 0, stream>>>(zbuf, p1W, p1b, z1, NG, ZK, DD, 1);
  dense<<<cdiv((long long)NG * (DD / 2), T), T, 0, stream>>>(z1, p2W, p2b, z2, NG, DD, DD / 2, 1);
  dense<<<cdiv((long long)NG * 5, T), T, 0, stream>>>(z2, p3W, p3b, (float*)d_out, NG, DD / 2, 5, 0);
}